// ChamferDistanceLoss_2241972929056
// MI455X (gfx1250) — compile-verified
//
#include <hip/hip_runtime.h>
#include <hip/hip_bf16.h>

// ---------------------------------------------------------------------------
// Chamfer distance, MI455X (gfx1250, wave32, WMMA).
//
// d2[i][j] = |x_i|^2 + |y_j|^2 - 2 x_i . y_j
// Cross term done with V_WMMA_F32_16X16X4_F32 (K padded 3 -> 4 with zeros).
// Trick: A is pre-scaled by -2 and the WMMA accumulator C is seeded with
// |y_j|^2, so the WMMA directly yields (|y_j|^2 - 2 x.y); |x_i|^2 is added
// after the min (row-constant), and sqrt is applied only to the 16K final
// mins (sqrt is monotone). Inner loop per 16x16 tile: 3 loads + |y|^2 +
// 8 v_mov (C seed) + 1 v_wmma + 8 v_min.
// ---------------------------------------------------------------------------

typedef float v2f __attribute__((ext_vector_type(2)));
typedef float v8f __attribute__((ext_vector_type(8)));

__global__ __launch_bounds__(256) void chamfer_dir_kernel(
    const float* __restrict__ X,   // [N,3] "query" points (rows of the tile)
    const float* __restrict__ Y,   // [M,3] "database" points (columns)
    int N, int M,
    float* __restrict__ out_dist)  // [N] min_j ||x_i - y_j||
{
    const int lane   = threadIdx.x & 31;
    const int wave   = threadIdx.x >> 5;
    const int i0     = (blockIdx.x * 8 + wave) * 16;   // wave-uniform
    if (i0 >= N) return;                               // uniform per wave

    const int  lo = lane & 15;
    const bool hi = lane >= 16;

    // ---- Load A tile (rows i0..i0+15), scaled by -2 --------------------
    // A 16x4 layout: lanes 0-15 hold K=0,1; lanes 16-31 hold K=2,3 (pad=0).
    int   m  = i0 + lo;  if (m > N - 1) m = N - 1;
    float x0 = X[3 * m], x1 = X[3 * m + 1], x2 = X[3 * m + 2];
    float xnorm = x0 * x0 + x1 * x1 + x2 * x2;
    v2f a;
    a.x = -2.0f * (hi ? x2 : x0);
    a.y =  hi ? 0.0f : (-2.0f * x1);

    // |x|^2 for the 8 rows this lane's accumulator covers (M = v + 8*hi).
    // Row r's norm lives in lane r (and r+16); gather while EXEC is full.
    float x2row[8];
#pragma unroll
    for (int v = 0; v < 8; ++v)
        x2row[v] = __shfl(xnorm, v + (hi ? 8 : 0), 32);

    float rmin[8];
#pragma unroll
    for (int v = 0; v < 8; ++v) rmin[v] = 3.402823466e38f;

    // ---- Sweep all column tiles ----------------------------------------
    for (int jt = 0; jt < M; jt += 16) {
        int   n  = jt + lo;  if (n > M - 1) n = M - 1;   // dup ok for min
        float y0 = Y[3 * n], y1 = Y[3 * n + 1], y2 = Y[3 * n + 2];
        float ynorm = y0 * y0 + y1 * y1 + y2 * y2;

        // B 4x16 layout mirrors A: lanes 0-15 hold K=0,1; 16-31 hold K=2,3.
        v2f b;
        b.x = hi ? y2 : y0;
        b.y = hi ? 0.0f : y1;

        // Seed C with |y_n|^2 (each lane owns column n = lane&15 for all v).
        v8f c;
#pragma unroll
        for (int v = 0; v < 8; ++v) c[v] = ynorm;

        // D = (-2X) * Y^T + |y|^2  ->  per-element (|y_n|^2 - 2 x_m.y_n)
        c = __builtin_amdgcn_wmma_f32_16x16x4_f32(
                /*neg_a=*/false, a, /*neg_b=*/false, b,
                /*c_mod=*/(short)0, c, /*reuse_a=*/false, /*reuse_b=*/false);

#pragma unroll
        for (int v = 0; v < 8; ++v) rmin[v] = fminf(rmin[v], c[v]);
    }

    // ---- Reduce across the 16 column lanes of each half ----------------
#pragma unroll
    for (int v = 0; v < 8; ++v) {
#pragma unroll
        for (int mask = 8; mask >= 1; mask >>= 1)
            rmin[v] = fminf(rmin[v], __shfl_xor(rmin[v], mask, 32));
    }

    if (lo == 0) {
#pragma unroll
        for (int v = 0; v < 8; ++v) {
            int row = i0 + v + (hi ? 8 : 0);
            if (row < N) {
                float d2 = x2row[v] + rmin[v];
                d2 = fmaxf(d2, 0.0f);
                out_dist[row] = sqrtf(d2);
            }
        }
    }
}

// ---------------------------------------------------------------------------
// Deterministic single-block reduction: mean(fwd) + mean(bwd) -> out[0].
// ---------------------------------------------------------------------------
__global__ __launch_bounds__(256) void chamfer_reduce_kernel(
    const float* __restrict__ fwd, int N,
    const float* __restrict__ bwd, int M,
    float* __restrict__ out)
{
    __shared__ float sdata[256];
    const float invN = 1.0f / (float)N;
    const float invM = 1.0f / (float)M;
    float acc = 0.0f;
    for (int i = threadIdx.x; i < N; i += 256) acc += fwd[i] * invN;
    for (int i = threadIdx.x; i < M; i += 256) acc += bwd[i] * invM;
    sdata[threadIdx.x] = acc;
    __syncthreads();
    for (int s = 128; s > 0; s >>= 1) {
        if (threadIdx.x < s) sdata[threadIdx.x] += sdata[threadIdx.x + s];
        __syncthreads();
    }
    if (threadIdx.x == 0) out[0] = sdata[0];
}

extern "C" void kernel_launch(void* const* d_in, const int* in_sizes, int n_in,
                              void* d_out, int out_size, void* d_ws, size_t ws_size,
                              hipStream_t stream)
{
    const float* X = (const float*)d_in[0];   // predicted_set [N,3] f32
    const float* Y = (const float*)d_in[1];   // target_set    [M,3] f32
    const int D = 3;
    const int N = in_sizes[0] / D;
    const int M = in_sizes[1] / D;

    float* fwd = (float*)d_ws;       // [N] per-point forward distances
    float* bwd = fwd + N;            // [M] per-point backward distances

    const int wavesF  = (N + 15) / 16;
    const int wavesB  = (M + 15) / 16;
    const int blocksF = (wavesF + 7) / 8;    // 8 waves (256 threads) / block
    const int blocksB = (wavesB + 7) / 8;

    chamfer_dir_kernel<<<blocksF, 256, 0, stream>>>(X, Y, N, M, fwd);
    chamfer_dir_kernel<<<blocksB, 256, 0, stream>>>(Y, X, M, N, bwd);
    chamfer_reduce_kernel<<<1, 256, 0, stream>>>(fwd, N, bwd, M, (float*)d_out);
}